// GroupedQueryAttention_86380382257377
// MI455X (gfx1250) — compile-verified
//
#include <hip/hip_runtime.h>
#include <hip/hip_bf16.h>

// Problem constants (match reference)
#define BB 2
#define SS 2048
#define DD 512
#define GG 8
#define HH 4
#define DG (DD * GG)        // 4096
#define DGH (DD * GG * HH)  // 16384

typedef __attribute__((ext_vector_type(2))) float v2f;
typedef __attribute__((ext_vector_type(8))) float v8f;

// CDNA5 async global->LDS copy (16B per lane), tracked by ASYNCcnt.
#define ASYNC_B128(ldsptr, gptr)                                      \
  asm volatile("global_load_async_to_lds_b128 %0, %1, off"            \
               :: "v"((unsigned)(uintptr_t)(ldsptr)),                 \
                  "v"((unsigned long long)(uintptr_t)(gptr))          \
               : "memory")
#define WAIT_ASYNC(n) asm volatile("s_wait_asynccnt " #n ::: "memory")

__device__ __forceinline__ float wave_reduce_sum(float v) {
  v += __shfl_xor(v, 1, 32);
  v += __shfl_xor(v, 2, 32);
  v += __shfl_xor(v, 4, 32);
  v += __shfl_xor(v, 8, 32);
  v += __shfl_xor(v, 16, 32);
  return v;
}

// K1a: xpart[b,chunk,d] = sum over 128 s of x[b, chunk*128+s, d]
__global__ __launch_bounds__(256) void k_colsum1(const float* __restrict__ x,
                                                 float* __restrict__ xpart) {
  int idx = blockIdx.x * 256 + threadIdx.x;  // 0..16383
  int b = idx >> 13;
  int chunk = (idx >> 9) & 15;
  int d = idx & (DD - 1);
  const float* p = x + ((size_t)b * SS + chunk * 128) * DD + d;
  float acc = 0.f;
  for (int s = 0; s < 128; ++s) acc += p[(size_t)s * DD];
  xpart[idx] = acc;
}

// K1b: xsum[b,d] = sum_chunk xpart[b,chunk,d]
__global__ __launch_bounds__(256) void k_colsum2(const float* __restrict__ xpart,
                                                 float* __restrict__ xsum) {
  int idx = blockIdx.x * 256 + threadIdx.x;  // 0..1023
  int b = idx >> 9, d = idx & (DD - 1);
  float acc = 0.f;
  for (int c = 0; c < 16; ++c) acc += xpart[(size_t)b * 8192 + c * DD + d];
  xsum[idx] = acc;
}

// K2: ksum[b,j] = xsum[b,:]@Wk[:,j] + S*bk[j]   (j = g*512+d)
__global__ __launch_bounds__(256) void k_ksum(const float* __restrict__ xsum,
                                              const float* __restrict__ Wk,
                                              const float* __restrict__ bk,
                                              float* __restrict__ ksum) {
  int j = blockIdx.x * 256 + threadIdx.x;  // 0..4095
  float a0 = 0.f, a1 = 0.f;
  for (int c = 0; c < DD; ++c) {
    float w = Wk[(size_t)c * DG + j];
    a0 += xsum[c] * w;
    a1 += xsum[DD + c] * w;
  }
  float bias = (float)SS * bk[j];
  ksum[j] = a0 + bias;
  ksum[DG + j] = a1 + bias;
}

// K3: wqeff[b, r, h*8+g] = sum_d Row_r[g*2048+h*512+d] * ksum[b, g*512+d]
//     rows r in [0,512) are Wq rows; r==512 is bq (bias row). Wave per row,
//     float4-vectorized over d (this kernel streams the 32 MB Wq once).
__global__ __launch_bounds__(256) void k_wqeff(const float* __restrict__ Wq,
                                               const float* __restrict__ bq,
                                               const float* __restrict__ ksum,
                                               float* __restrict__ wqeff) {
  int wave = blockIdx.x * 8 + (threadIdx.x >> 5);
  int lane = threadIdx.x & 31;
  if (wave > DD) return;  // rows 0..512 inclusive
  const float* arow = (wave < DD) ? (Wq + (size_t)wave * DGH) : bq;
  for (int t2 = 0; t2 < 32; ++t2) {  // segment t2 = g*4+h (contiguous 512 floats)
    int g = t2 >> 2, h = t2 & 3;
    const float4* seg4 = (const float4*)(arow + t2 * DD);
    const float4* k04 = (const float4*)(ksum + g * DD);
    const float4* k14 = (const float4*)(ksum + DG + g * DD);
    float a0 = 0.f, a1 = 0.f;
    for (int i = lane; i < DD / 4; i += 32) {
      float4 w = seg4[i];
      float4 u = k04[i];
      float4 v = k14[i];
      a0 += w.x * u.x + w.y * u.y + w.z * u.z + w.w * u.w;
      a1 += w.x * v.x + w.y * v.y + w.z * v.z + w.w * v.w;
    }
    a0 = wave_reduce_sum(a0);
    a1 = wave_reduce_sum(a1);
    if (lane == 0) {
      int hg = h * GG + g;
      wqeff[((size_t)0 * 513 + wave) * 32 + hg] = a0;
      wqeff[((size_t)1 * 513 + wave) * 32 + hg] = a1;
    }
  }
}

// K4: per (b,q) wave: scores[hg] = (x[b,q,:]@wqeff[b,:,hg] + bias_row)/sqrt(D);
//     softmax over g (groups of 8 lanes, L = h*8+g); part[b,q,g] = sum_h weights
__global__ __launch_bounds__(256) void k_scores(const float* __restrict__ x,
                                                const float* __restrict__ wqeff,
                                                float* __restrict__ part) {
  int gw = blockIdx.x * 8 + (threadIdx.x >> 5);  // 0..4095 (b*2048+q)
  int lane = threadIdx.x & 31;
  int b = gw >> 11;
  int q = gw & (SS - 1);
  const float* xrow = x + ((size_t)b * SS + q) * DD;
  const float* wcol = wqeff + (size_t)b * 513 * 32 + lane;
  float acc = wcol[(size_t)DD * 32];  // bias row (r = 512)
  for (int c = 0; c < DD; ++c) acc += xrow[c] * wcol[(size_t)c * 32];
  const float INV_SQRT_D = 0.04419417382415922f;  // 1/sqrt(512)
  float score = acc * INV_SQRT_D;
  float m = score;
  m = fmaxf(m, __shfl_xor(m, 1, 32));
  m = fmaxf(m, __shfl_xor(m, 2, 32));
  m = fmaxf(m, __shfl_xor(m, 4, 32));
  float e = __expf(score - m);
  float s = e;
  s += __shfl_xor(s, 1, 32);
  s += __shfl_xor(s, 2, 32);
  s += __shfl_xor(s, 4, 32);
  float w = e / s;
  w += __shfl_xor(w, 8, 32);   // sum over h
  w += __shfl_xor(w, 16, 32);
  if (lane < 8) part[((size_t)b * SS + q) * 8 + lane] = w;
}

// K5: wsum[b*8+g] = sum_q part[b,q,g]  (deterministic two-stage reduce)
__global__ __launch_bounds__(256) void k_wsum(const float* __restrict__ part,
                                              float* __restrict__ wsum) {
  __shared__ float lds[256];
  int t = threadIdx.x;
  int bg = t & 15;
  int chunk = t >> 4;
  int b = bg >> 3, g = bg & 7;
  float acc = 0.f;
  for (int q = chunk * 128; q < chunk * 128 + 128; ++q)
    acc += part[((size_t)b * SS + q) * 8 + g];
  lds[t] = acc;
  __syncthreads();
  if (t < 16) {
    float s = 0.f;
    for (int i = 0; i < 16; ++i) s += lds[i * 16 + t];
    wsum[t] = s;
  }
}

// K5b: bvwo[g,e] = sum_d bv[g*512+d] * Wo[g*512+d, e]
__global__ __launch_bounds__(256) void k_bvwo(const float* __restrict__ bv,
                                              const float* __restrict__ Wo,
                                              float* __restrict__ bvwo) {
  int idx = blockIdx.x * 256 + threadIdx.x;  // 0..4095
  int g = idx >> 9, e = idx & (DD - 1);
  const float* col = Wo + (size_t)(g * DD) * DD + e;
  const float* bvg = bv + g * DD;
  float acc = 0.f;
  for (int d = 0; d < DD; ++d) acc += bvg[d] * col[(size_t)d * DD];
  bvwo[idx] = acc;
}

// ---- WMMA GEMM kernels -----------------------------------------------------
// Block = 256 threads (8 waves as 2x4 grid of 16x16 fp32 WMMA tiles).
// Block tile 32(M) x 64(N), K step 16, double-buffered async global->LDS.
// LDS row strides chosen 16B-aligned (b128) and bank-conflict-free:
//   As stride 20 floats, Bs stride 72 floats.

// K6: Weff[b] = (Wv with column-block g scaled by wsum[b,g]) @ Wo
//     M=512, N=512, K=4096; both batches accumulated in one pass.
//     Stage loop is split g-major so the per-g scale pair is loaded once
//     per 32 stages (keeps the inner WMMA loop free of VMEM waits).
__global__ __launch_bounds__(256) void k_weff(const float* __restrict__ Wv,
                                              const float* __restrict__ Wo,
                                              const float* __restrict__ wsum,
                                              float* __restrict__ weff) {
  __shared__ float As[2][32][20];
  __shared__ float Bs[2][16][72];
  const int tid = threadIdx.x;
  const int lane = tid & 31;
  const int wv = tid >> 5;
  const int mw = (wv >> 2) * 16;
  const int nw = (wv & 3) * 16;
  const int rowBase = blockIdx.y * 32;
  const int colBase = blockIdx.x * 64;
  const int mrow = lane & 15;
  const int kh = (lane >> 4) * 2;  // ISA 7.12.2: K split across lane halves

  // Async transfer coords: every wave issues exactly 2 b128 ops per stage
  // (A xfers duplicated across lane halves -> EXEC stays full, uniform cnt).
  const int ai = wv * 16 + (lane & 15);        // 0..127 (x2 duplication)
  const int am = ai >> 2, ak = (ai & 3) * 4;   // A tile: row am, cols ak..ak+3
  const int bi = wv * 32 + lane;               // 0..255
  const int bk = bi >> 4, be = (bi & 15) * 4;  // B tile: row bk, cols be..be+3
  const float* gA = Wv + (size_t)(rowBase + am) * DG + ak;
  const float* gB = Wo + (size_t)bk * DD + colBase + be;

  v8f c0 = {};
  v8f c1 = {};

  ASYNC_B128(&As[0][am][ak], gA);
  ASYNC_B128(&Bs[0][bk][be], gB);

  const int NS = DG / 16;  // 256 stages total; 32 per g-block
  for (int g = 0; g < GG; ++g) {
    const float sc0 = wsum[g];
    const float sc1 = wsum[GG + g];
    for (int t = 0; t < 32; ++t) {
      const int s = g * 32 + t;
      const int cur = s & 1;
      if (s + 1 < NS) {  // uniform across block
        const int kNext = (s + 1) * 16;
        ASYNC_B128(&As[1 - cur][am][ak], gA + kNext);
        ASYNC_B128(&Bs[1 - cur][bk][be], gB + (size_t)kNext * DD);
        WAIT_ASYNC(0x2);  // current stage done; next stage in flight
      } else {
        WAIT_ASYNC(0x0);
      }
      __syncthreads();
      for (int ks = 0; ks < 16; ks += 4) {
        v2f a, bf;
        a.x = As[cur][mw + mrow][ks + kh];
        a.y = As[cur][mw + mrow][ks + kh + 1];
        bf.x = Bs[cur][ks + kh][nw + mrow];
        bf.y = Bs[cur][ks + kh + 1][nw + mrow];
        v2f a0 = a * sc0;
        v2f a1 = a * sc1;
        c0 = __builtin_amdgcn_wmma_f32_16x16x4_f32(false, a0, false, bf, (short)0, c0, false, false);
        c1 = __builtin_amdgcn_wmma_f32_16x16x4_f32(false, a1, false, bf, (short)0, c1, false, false);
      }
      __syncthreads();
    }
  }
  int mOut = rowBase + mw + (lane >> 4) * 8;  // C: VGPR r <-> rows r / r+8
  int col = colBase + nw + (lane & 15);
  for (int r = 0; r < 8; ++r) {
    weff[(size_t)(mOut + r) * DD + col] = c0[r];
    weff[(size_t)DD * DD + (size_t)(mOut + r) * DD + col] = c1[r];
  }
}

// K7: out[b*S+s, :] = x[b,s,:] @ Weff[b] + (bo + sum_g wsum[b,g]*bvwo[g,:])
//     M=4096, N=512, K=512.
__global__ __launch_bounds__(256) void k_out(const float* __restrict__ x,
                                             const float* __restrict__ weff,
                                             const float* __restrict__ wsum,
                                             const float* __restrict__ bvwo,
                                             const float* __restrict__ bo,
                                             float* __restrict__ out) {
  __shared__ float As[2][32][20];
  __shared__ float Bs[2][16][72];
  const int tid = threadIdx.x;
  const int lane = tid & 31;
  const int wv = tid >> 5;
  const int mw = (wv >> 2) * 16;
  const int nw = (wv & 3) * 16;
  const int rowBase = blockIdx.y * 32;  // over b*S (tiles never straddle batches)
  const int colBase = blockIdx.x * 64;
  const int b = rowBase >> 11;
  const float* Bm = weff + (size_t)b * DD * DD;
  const int mrow = lane & 15;
  const int kh = (lane >> 4) * 2;

  const int ai = wv * 16 + (lane & 15);
  const int am = ai >> 2, ak = (ai & 3) * 4;
  const int bi = wv * 32 + lane;
  const int bk = bi >> 4, be = (bi & 15) * 4;
  const float* gA = x + (size_t)(rowBase + am) * DD + ak;
  const float* gB = Bm + (size_t)bk * DD + colBase + be;

  v8f c0 = {};

  ASYNC_B128(&As[0][am][ak], gA);
  ASYNC_B128(&Bs[0][bk][be], gB);

  const int NS = DD / 16;  // 32 stages
  for (int s = 0; s < NS; ++s) {
    const int cur = s & 1;
    if (s + 1 < NS) {
      const int kNext = (s + 1) * 16;
      ASYNC_B128(&As[1 - cur][am][ak], gA + kNext);
      ASYNC_B128(&Bs[1 - cur][bk][be], gB + (size_t)kNext * DD);
      WAIT_ASYNC(0x2);
    } else {
      WAIT_ASYNC(0x0);
    }
    __syncthreads();
    for (int ks = 0; ks < 16; ks += 4) {
      v2f a, bf;
      a.x = As[cur][mw + mrow][ks + kh];
      a.y = As[cur][mw + mrow][ks + kh + 1];
      bf.x = Bs[cur][ks + kh][nw + mrow];
      bf.y = Bs[cur][ks + kh + 1][nw + mrow];
      c0 = __builtin_amdgcn_wmma_f32_16x16x4_f32(false, a, false, bf, (short)0, c0, false, false);
    }
    __syncthreads();
  }
  int col = colBase + nw + (lane & 15);
  float cb = bo[col];
  for (int g = 0; g < GG; ++g) cb += wsum[b * GG + g] * bvwo[g * DD + col];
  int mOut = rowBase + mw + (lane >> 4) * 8;
  for (int r = 0; r < 8; ++r)
    out[(size_t)(mOut + r) * DD + col] = c0[r] + cb;
}

extern "C" void kernel_launch(void* const* d_in, const int* in_sizes, int n_in,
                              void* d_out, int out_size, void* d_ws, size_t ws_size,
                              hipStream_t stream) {
  const float* x  = (const float*)d_in[0];
  const float* Wq = (const float*)d_in[1];
  const float* bq = (const float*)d_in[2];
  const float* Wk = (const float*)d_in[3];
  const float* bk = (const float*)d_in[4];
  const float* Wv = (const float*)d_in[5];
  const float* bv = (const float*)d_in[6];
  const float* Wo = (const float*)d_in[7];
  const float* bo = (const float*)d_in[8];
  float* out = (float*)d_out;

  // Workspace layout (floats): ~620k floats (~2.5 MB)
  float* ws    = (float*)d_ws;
  float* xpart = ws;              // 2*16*512 = 16384
  float* xsum  = xpart + 16384;   // 1024
  float* ksum  = xsum + 1024;     // 8192
  float* wqeff = ksum + 8192;     // 2*513*32 = 32832
  float* part  = wqeff + 32832;   // 2*2048*8 = 32768
  float* wsum  = part + 32768;    // 16
  float* bvwo  = wsum + 16;       // 8*512 = 4096
  float* weff  = bvwo + 4096;     // 2*512*512 = 524288

  k_colsum1<<<64, 256, 0, stream>>>(x, xpart);
  k_colsum2<<<4, 256, 0, stream>>>(xpart, xsum);
  k_ksum<<<16, 256, 0, stream>>>(xsum, Wk, bk, ksum);
  k_wqeff<<<65, 256, 0, stream>>>(Wq, bq, ksum, wqeff);
  k_scores<<<512, 256, 0, stream>>>(x, wqeff, part);
  k_wsum<<<1, 256, 0, stream>>>(part, wsum);
  k_bvwo<<<16, 256, 0, stream>>>(bv, Wo, bvwo);
  k_weff<<<dim3(8, 16), 256, 0, stream>>>(Wv, Wo, wsum, weff);
  k_out<<<dim3(8, 128), 256, 0, stream>>>(x, weff, wsum, bvwo, bo, out);
}